// ResD1LayerAdaptiveK_6176162972403
// MI455X (gfx1250) — compile-verified
//
#include <hip/hip_runtime.h>
#include <math.h>

#define B_N 768
#define D_N 256
#define H_N 256
#define KMAX 16

typedef float v2f __attribute__((ext_vector_type(2)));
typedef float v8f __attribute__((ext_vector_type(8)));

// ---------------------------------------------------------------------------
// 0) init device-side scalars (ws is poisoned; must be deterministic per call)
// ---------------------------------------------------------------------------
__global__ void init_ws_kernel(int* maxk) {
    if (threadIdx.x == 0) *maxk = 0;
}

// ---------------------------------------------------------------------------
// 1) per-row stats: tau_i (softplus path), k_i, ||x_i||^2, atomicMax(max_k)
// ---------------------------------------------------------------------------
__global__ void row_stats_kernel(const float* __restrict__ x,
                                 const float* __restrict__ W_tau,
                                 const float* __restrict__ b_tau,
                                 float* __restrict__ tau,
                                 int* __restrict__ kvals,
                                 float* __restrict__ norms,
                                 int* __restrict__ maxk) {
    __shared__ float s0[D_N];
    __shared__ float s1[D_N];
    const int i = blockIdx.x, t = threadIdx.x;
    float xv = x[i * D_N + t];
    s0[t] = xv * W_tau[t];
    s1[t] = xv * xv;
    __syncthreads();
    for (int off = D_N / 2; off > 0; off >>= 1) {
        if (t < off) { s0[t] += s0[t + off]; s1[t] += s1[t + off]; }
        __syncthreads();
    }
    if (t == 0) {
        float v  = s0[0] + b_tau[0];
        float sp = (v > 20.0f) ? v : log1pf(expf(v));   // softplus
        float tv = fmaxf(sp, 0.01f) + 1.0f;
        tau[i] = tv;
        int k = (int)floorf((float)KMAX * expf(-0.5f * tv));
        k = k < 1 ? 1 : (k > B_N ? B_N : k);
        kvals[i] = k;
        atomicMax(maxk, k);
        norms[i] = s1[0];
    }
}

// ---------------------------------------------------------------------------
// 2) transpose the three 256x256 weight matrices so GEMM B-fragments become
//    K-contiguous b64 loads (same pattern the Gram kernel enjoys for free)
// ---------------------------------------------------------------------------
__global__ void transpose256_kernel(const float* __restrict__ A0,
                                    const float* __restrict__ A1,
                                    const float* __restrict__ A2,
                                    float* __restrict__ T0,
                                    float* __restrict__ T1,
                                    float* __restrict__ T2) {
    __shared__ float tile[32][33];
    const float* A; float* T;
    if (blockIdx.z == 0)      { A = A0; T = T0; }
    else if (blockIdx.z == 1) { A = A1; T = T1; }
    else                      { A = A2; T = T2; }
    const int bx = blockIdx.x * 32, by = blockIdx.y * 32;
    const int tx = threadIdx.x & 31, ty = threadIdx.x >> 5;   // 256 thr = 8 rows/pass
#pragma unroll
    for (int r = ty; r < 32; r += 8)
        tile[r][tx] = A[(size_t)(by + r) * H_N + bx + tx];
    __syncthreads();
#pragma unroll
    for (int r = ty; r < 32; r += 8)
        T[(size_t)(bx + r) * D_N + by + tx] = tile[tx][r];
}

// ---------------------------------------------------------------------------
// 3) Gram via WMMA f32 16x16x4: dist2[i,j] = ||xi||^2 + ||xj||^2 - 2 xi.xj
//    (sqrt skipped: monotone for top-k ordering)
// ---------------------------------------------------------------------------
__global__ void gram_dist_kernel(const float* __restrict__ x,
                                 const float* __restrict__ norms,
                                 float* __restrict__ dist2) {
    const int tilesN = B_N / 16;                 // 48
    const int wave = threadIdx.x >> 5;
    const int lane = threadIdx.x & 31;
    const int tile = blockIdx.x * (blockDim.x >> 5) + wave;
    const int tm = tile / tilesN, tn = tile % tilesN;

    const float* aRow = x + (size_t)(tm * 16 + (lane & 15)) * D_N;
    const float* bRow = x + (size_t)(tn * 16 + (lane & 15)) * D_N;  // B = x^T -> contiguous
    const int kOff = (lane >> 4) * 2;
    __builtin_prefetch(aRow, 0, 1);
    __builtin_prefetch(bRow, 0, 1);

    v8f acc = {0.f, 0.f, 0.f, 0.f, 0.f, 0.f, 0.f, 0.f};
    for (int k0 = 0; k0 < D_N; k0 += 4) {
        v2f af = *(const v2f*)(aRow + k0 + kOff);
        v2f bf = *(const v2f*)(bRow + k0 + kOff);
        acc = __builtin_amdgcn_wmma_f32_16x16x4_f32(false, af, false, bf,
                                                    (short)0, acc, false, false);
    }
    const int col = tn * 16 + (lane & 15);
    const float ncol = norms[col];
#pragma unroll
    for (int v = 0; v < 8; v++) {
        int row = tm * 16 + v + ((lane >> 4) << 3);
        float d = norms[row] + ncol - 2.0f * acc[v];
        dist2[(size_t)row * B_N + col] = fmaxf(d, 0.0f);
    }
}

// ---------------------------------------------------------------------------
// 4) per-row top-16 smallest distances (ties -> lower index, like lax.top_k)
//    one wave32 per row, iterative selection + butterfly reduction
// ---------------------------------------------------------------------------
__global__ void topk_kernel(const float* __restrict__ dist2,
                            int* __restrict__ topk) {
    const int i = blockIdx.x;
    const int lane = threadIdx.x;                // blockDim = 32
    const float* row = dist2 + (size_t)i * B_N;
    int chosen[KMAX];
#pragma unroll
    for (int p = 0; p < KMAX; p++) {
        float bv = INFINITY; int bi = B_N;
        for (int c = lane; c < B_N; c += 32) {
            bool excl = false;
            for (int q = 0; q < p; q++) excl |= (chosen[q] == c);
            float v = row[c];
            if (!excl && v < bv) { bv = v; bi = c; }
        }
        for (int m = 16; m > 0; m >>= 1) {       // wave32 butterfly min-reduce
            float ov = __shfl_xor(bv, m, 32);
            int   oi = __shfl_xor(bi, m, 32);
            if (ov < bv || (ov == bv && oi < bi)) { bv = ov; bi = oi; }
        }
        chosen[p] = bi;
        if (lane == 0) topk[i * KMAX + p] = bi;
    }
}

// ---------------------------------------------------------------------------
// 5) S[i,:] = sum_{j<max_k} |x_i - x_sel|, sel = topk[i][j] if j<k_i else 0
// ---------------------------------------------------------------------------
__global__ void neighbor_sum_kernel(const float* __restrict__ x,
                                    const int* __restrict__ topk,
                                    const int* __restrict__ kvals,
                                    const int* __restrict__ maxk,
                                    float* __restrict__ S) {
    __shared__ int sel[KMAX];
    const int i = blockIdx.x, t = threadIdx.x;
    const int mk = *maxk;
    const int ki = kvals[i];
    if (t < KMAX) sel[t] = (t < ki) ? topk[i * KMAX + t] : 0;
    __syncthreads();
    float xi = x[i * D_N + t];
    float acc = 0.0f;
    for (int j = 0; j < mk; j++)
        acc += fabsf(xi - x[(size_t)sel[j] * D_N + t]);
    S[i * D_N + t] = acc;
}

// ---------------------------------------------------------------------------
// 6) h_k = (S @ W_diff / max_k + b_diff) / tau     (WMMA f32, Wt = W_diff^T)
// ---------------------------------------------------------------------------
__global__ void gemm_proj_kernel(const float* __restrict__ S,
                                 const float* __restrict__ Wt_diff,  // [H,D]
                                 const float* __restrict__ b_diff,
                                 const float* __restrict__ tau,
                                 const int* __restrict__ maxk,
                                 float* __restrict__ hk) {
    const int tilesN = H_N / 16;                 // 16
    const int wave = threadIdx.x >> 5;
    const int lane = threadIdx.x & 31;
    const int tile = blockIdx.x * (blockDim.x >> 5) + wave;
    const int tm = tile / tilesN, tn = tile % tilesN;

    const float* aRow = S + (size_t)(tm * 16 + (lane & 15)) * D_N;
    const float* bRow = Wt_diff + (size_t)(tn * 16 + (lane & 15)) * D_N;
    const int kOff = (lane >> 4) * 2;
    __builtin_prefetch(aRow, 0, 1);

    v8f acc = {0.f, 0.f, 0.f, 0.f, 0.f, 0.f, 0.f, 0.f};
    for (int k0 = 0; k0 < D_N; k0 += 4) {
        v2f af = *(const v2f*)(aRow + k0 + kOff);
        v2f bf = *(const v2f*)(bRow + k0 + kOff);
        acc = __builtin_amdgcn_wmma_f32_16x16x4_f32(false, af, false, bf,
                                                    (short)0, acc, false, false);
    }
    const float rmk = 1.0f / (float)(*maxk);
    const int col = tn * 16 + (lane & 15);
    const float bd = b_diff[col];
#pragma unroll
    for (int v = 0; v < 8; v++) {
        int row = tm * 16 + v + ((lane >> 4) << 3);
        hk[(size_t)row * H_N + col] = (acc[v] * rmk + bd) / tau[row];
    }
}

// ---------------------------------------------------------------------------
// 7) z = relu(h_k @ W_agg + b_agg) + x @ W_res + b_res   (2x WMMA per wave)
// ---------------------------------------------------------------------------
__global__ void gemm_fused_kernel(const float* __restrict__ hk,
                                  const float* __restrict__ x,
                                  const float* __restrict__ Wt_agg,   // [H,H]
                                  const float* __restrict__ b_agg,
                                  const float* __restrict__ Wt_res,   // [H,D]
                                  const float* __restrict__ b_res,
                                  float* __restrict__ z) {
    const int tilesN = H_N / 16;
    const int wave = threadIdx.x >> 5;
    const int lane = threadIdx.x & 31;
    const int tile = blockIdx.x * (blockDim.x >> 5) + wave;
    const int tm = tile / tilesN, tn = tile % tilesN;

    const float* a1Row = hk + (size_t)(tm * 16 + (lane & 15)) * H_N;
    const float* a2Row = x  + (size_t)(tm * 16 + (lane & 15)) * D_N;
    const float* b1Row = Wt_agg + (size_t)(tn * 16 + (lane & 15)) * H_N;
    const float* b2Row = Wt_res + (size_t)(tn * 16 + (lane & 15)) * D_N;
    const int kOff = (lane >> 4) * 2;
    __builtin_prefetch(a1Row, 0, 1);
    __builtin_prefetch(a2Row, 0, 1);

    v8f acc1 = {0.f, 0.f, 0.f, 0.f, 0.f, 0.f, 0.f, 0.f};
    v8f acc2 = {0.f, 0.f, 0.f, 0.f, 0.f, 0.f, 0.f, 0.f};
    for (int k0 = 0; k0 < D_N; k0 += 4) {
        v2f a1 = *(const v2f*)(a1Row + k0 + kOff);
        v2f b1 = *(const v2f*)(b1Row + k0 + kOff);
        v2f a2 = *(const v2f*)(a2Row + k0 + kOff);
        v2f b2 = *(const v2f*)(b2Row + k0 + kOff);
        acc1 = __builtin_amdgcn_wmma_f32_16x16x4_f32(false, a1, false, b1,
                                                     (short)0, acc1, false, false);
        acc2 = __builtin_amdgcn_wmma_f32_16x16x4_f32(false, a2, false, b2,
                                                     (short)0, acc2, false, false);
    }
    const int col = tn * 16 + (lane & 15);
    const float ba = b_agg[col], br = b_res[col];
#pragma unroll
    for (int v = 0; v < 8; v++) {
        int row = tm * 16 + v + ((lane >> 4) << 3);
        float h = acc1[v] + ba;
        h = h > 0.0f ? h : 0.0f;                 // relu
        z[(size_t)row * H_N + col] = h + acc2[v] + br;
    }
}

// ---------------------------------------------------------------------------
// 8) LayerNorm (population var, eps=1e-5)
// ---------------------------------------------------------------------------
__global__ void layernorm_kernel(const float* __restrict__ z,
                                 const float* __restrict__ gamma,
                                 const float* __restrict__ beta,
                                 float* __restrict__ out) {
    __shared__ float s0[H_N];
    __shared__ float s1[H_N];
    const int i = blockIdx.x, t = threadIdx.x;
    float v = z[i * H_N + t];
    s0[t] = v;
    s1[t] = v * v;
    __syncthreads();
    for (int off = H_N / 2; off > 0; off >>= 1) {
        if (t < off) { s0[t] += s0[t + off]; s1[t] += s1[t + off]; }
        __syncthreads();
    }
    float mean = s0[0] * (1.0f / H_N);
    float var  = s1[0] * (1.0f / H_N) - mean * mean;
    var = fmaxf(var, 0.0f);
    float inv = rsqrtf(var + 1e-5f);
    out[i * H_N + t] = (v - mean) * inv * gamma[t] + beta[t];
}

// ---------------------------------------------------------------------------
extern "C" void kernel_launch(void* const* d_in, const int* in_sizes, int n_in,
                              void* d_out, int out_size, void* d_ws, size_t ws_size,
                              hipStream_t stream) {
    (void)in_sizes; (void)n_in; (void)out_size; (void)ws_size;
    const float* x      = (const float*)d_in[0];
    const float* W_diff = (const float*)d_in[1];
    const float* b_diff = (const float*)d_in[2];
    const float* W_tau  = (const float*)d_in[3];
    const float* b_tau  = (const float*)d_in[4];
    const float* W_agg  = (const float*)d_in[5];
    const float* b_agg  = (const float*)d_in[6];
    const float* W_res  = (const float*)d_in[7];
    const float* b_res  = (const float*)d_in[8];
    const float* gamma  = (const float*)d_in[9];
    const float* beta   = (const float*)d_in[10];
    float* out = (float*)d_out;

    // workspace layout (all 4B aligned), ~5.6 MB total
    char* ws = (char*)d_ws;
    float* dist2   = (float*)ws;                                // 768*768
    int*   topk    = (int*)(ws + (size_t)B_N * B_N * 4);        // 768*16
    float* S       = (float*)((char*)topk + (size_t)B_N * KMAX * 4);  // 768*256
    float* hk      = S  + (size_t)B_N * D_N;                    // 768*256
    float* z       = hk + (size_t)B_N * H_N;                    // 768*256
    float* Wt_diff = z  + (size_t)B_N * H_N;                    // 256*256
    float* Wt_agg  = Wt_diff + (size_t)D_N * H_N;               // 256*256
    float* Wt_res  = Wt_agg  + (size_t)H_N * H_N;               // 256*256
    float* tau     = Wt_res  + (size_t)D_N * H_N;               // 768
    float* norms   = tau + B_N;                                 // 768
    int*   kvals   = (int*)(norms + B_N);                       // 768
    int*   maxk    = kvals + B_N;                               // 1

    init_ws_kernel<<<1, 1, 0, stream>>>(maxk);
    row_stats_kernel<<<B_N, D_N, 0, stream>>>(x, W_tau, b_tau, tau, kvals, norms, maxk);
    transpose256_kernel<<<dim3(8, 8, 3), 256, 0, stream>>>(W_diff, W_agg, W_res,
                                                           Wt_diff, Wt_agg, Wt_res);
    {   // Gram: (768/16)*(768/16) = 2304 tiles, 8 waves/block
        int tiles = (B_N / 16) * (B_N / 16);
        gram_dist_kernel<<<tiles / 8, 256, 0, stream>>>(x, norms, dist2);
    }
    topk_kernel<<<B_N, 32, 0, stream>>>(dist2, topk);
    neighbor_sum_kernel<<<B_N, D_N, 0, stream>>>(x, topk, kvals, maxk, S);
    {   // proj/fused: (768/16)*(256/16) = 768 tiles
        int tiles = (B_N / 16) * (H_N / 16);
        gemm_proj_kernel<<<tiles / 8, 256, 0, stream>>>(S, Wt_diff, b_diff, tau, maxk, hk);
        gemm_fused_kernel<<<tiles / 8, 256, 0, stream>>>(hk, x, Wt_agg, b_agg, Wt_res, b_res, z);
    }
    layernorm_kernel<<<B_N, H_N, 0, stream>>>(z, gamma, beta, out);
}